// GlobalPool_5119601016902
// MI455X (gfx1250) — compile-verified
//
#include <hip/hip_runtime.h>
#include <math.h>

typedef __attribute__((ext_vector_type(2))) float v2f;
typedef __attribute__((ext_vector_type(8))) float v8f;

// ---------------------------------------------------------------------------
// Kernel 1: fused gather + per-head attention logits + segment softmax +
// weighted segment sum.  One workgroup (256 thr = 8 waves) per graph b.
// Wave handles one node per iteration; lane l owns row elements [8l, 8l+8).
// Pass A: z stats (max per head).  Pass B: acc += e*hv, denom += e.
// he[b,:] = acc / max(denom, 1e-12).  (empty segments -> he = 0, matching ref)
// ---------------------------------------------------------------------------
__device__ __forceinline__ int lower_bound_i32(const int* __restrict__ a, int n, int key) {
    int lo = 0, hi = n;
    while (lo < hi) {
        int mid = (lo + hi) >> 1;
        if (a[mid] < key) lo = mid + 1; else hi = mid;
    }
    return lo;
}

__global__ __launch_bounds__(256) void pool_kernel(
    const float* __restrict__ node_feats,  // [N,256]
    const float* __restrict__ g_feats,     // [B,256]
    const float* __restrict__ degree,      // [N]
    const int*   __restrict__ seg,         // [N] sorted
    const float* __restrict__ attn,        // [256] = [4 heads][64]
    float* __restrict__ he_out,            // [B,256] (workspace)
    int N)
{
    __shared__ float lds_red[256];
    __shared__ float lds_acc[8 * 256];
    __shared__ float lds_denh[4];
    __shared__ int   lds_range[2];

    const int b    = blockIdx.x;
    const int tid  = threadIdx.x;
    const int lane = tid & 31;
    const int wave = tid >> 5;     // 0..7

    if (tid == 0) {
        lds_range[0] = lower_bound_i32(seg, N, b);
        lds_range[1] = lower_bound_i32(seg, N, b + 1);
    }
    __syncthreads();
    const int rs = lds_range[0];
    const int re = lds_range[1];

    // per-lane constants: graph residual slice + attn slice (8 floats each)
    float gv[8], av[8];
    {
        const float4* gp = (const float4*)(g_feats + (size_t)b * 256 + lane * 8);
        const float4* ap = (const float4*)(attn + lane * 8);
        float4 g0 = gp[0], g1 = gp[1], a0 = ap[0], a1 = ap[1];
        gv[0]=g0.x; gv[1]=g0.y; gv[2]=g0.z; gv[3]=g0.w;
        gv[4]=g1.x; gv[5]=g1.y; gv[6]=g1.z; gv[7]=g1.w;
        av[0]=a0.x; av[1]=a0.y; av[2]=a0.z; av[3]=a0.w;
        av[4]=a1.x; av[5]=a1.y; av[6]=a1.z; av[7]=a1.w;
    }

    // -------- Pass A: per-head max of z ------------------------------------
    float zmax_l = -__builtin_inff();
    for (int n = rs + wave; n < re; n += 8) {
        const float4* np = (const float4*)(node_feats + (size_t)n * 256 + lane * 8);
        float4 x0 = np[0], x1 = np[1];
        float hv[8];
        hv[0]=x0.x+gv[0]; hv[1]=x0.y+gv[1]; hv[2]=x0.z+gv[2]; hv[3]=x0.w+gv[3];
        hv[4]=x1.x+gv[4]; hv[5]=x1.y+gv[5]; hv[6]=x1.z+gv[6]; hv[7]=x1.w+gv[7];
        float part = 0.f;
        #pragma unroll
        for (int k = 0; k < 8; ++k) part = fmaf(hv[k], av[k], part);
        // reduce over the 8-lane head group (xor 1,2,4 stays inside group)
        part += __shfl_xor(part, 1);
        part += __shfl_xor(part, 2);
        part += __shfl_xor(part, 4);
        float z = part * degree[n];
        zmax_l = fmaxf(zmax_l, z);
    }
    lds_red[tid] = zmax_l;
    __syncthreads();
    float zmax = -__builtin_inff();
    #pragma unroll
    for (int w = 0; w < 8; ++w) zmax = fmaxf(zmax, lds_red[w * 32 + lane]);
    __syncthreads();   // lds_red reused below

    // -------- Pass B: acc += e*hv, denom += e (segment re-read hits L2) ----
    float acc[8];
    #pragma unroll
    for (int k = 0; k < 8; ++k) acc[k] = 0.f;
    float denom_l = 0.f;
    for (int n = rs + wave; n < re; n += 8) {
        const float4* np = (const float4*)(node_feats + (size_t)n * 256 + lane * 8);
        float4 x0 = np[0], x1 = np[1];
        float hv[8];
        hv[0]=x0.x+gv[0]; hv[1]=x0.y+gv[1]; hv[2]=x0.z+gv[2]; hv[3]=x0.w+gv[3];
        hv[4]=x1.x+gv[4]; hv[5]=x1.y+gv[5]; hv[6]=x1.z+gv[6]; hv[7]=x1.w+gv[7];
        float part = 0.f;
        #pragma unroll
        for (int k = 0; k < 8; ++k) part = fmaf(hv[k], av[k], part);
        part += __shfl_xor(part, 1);
        part += __shfl_xor(part, 2);
        part += __shfl_xor(part, 4);
        float z = part * degree[n];
        float e = __expf(z - zmax);
        denom_l += e;
        #pragma unroll
        for (int k = 0; k < 8; ++k) acc[k] = fmaf(e, hv[k], acc[k]);
    }

    lds_red[tid] = denom_l;
    #pragma unroll
    for (int k = 0; k < 8; ++k) lds_acc[wave * 256 + lane * 8 + k] = acc[k];
    __syncthreads();

    if (tid < 32) {
        float ds = 0.f;
        #pragma unroll
        for (int w = 0; w < 8; ++w) ds += lds_red[w * 32 + tid];
        if ((tid & 7) == 0) lds_denh[tid >> 3] = ds;   // one value per head
    }
    __syncthreads();

    float s = 0.f;
    #pragma unroll
    for (int w = 0; w < 8; ++w) s += lds_acc[w * 256 + tid];
    he_out[(size_t)b * 256 + tid] = s / fmaxf(lds_denh[tid >> 6], 1e-12f);
}

// ---------------------------------------------------------------------------
// Kernel 2: out = relu(he@W1^T + b1)@W2^T + b2 + g_feats, via fp32 WMMA.
// One workgroup per 16 output rows.  8 waves x 2 N-tiles each.
// LDS tiles padded to stride 260 floats -> conflict-free b64 reads.
// ---------------------------------------------------------------------------
#define LDS_STRIDE 260

__global__ __launch_bounds__(256) void mlp_kernel(
    const float* __restrict__ he,       // [B,256]
    const float* __restrict__ W1,       // [256,256] row-major (out, in)
    const float* __restrict__ b1,       // [256]
    const float* __restrict__ W2,       // [256,256]
    const float* __restrict__ b2,       // [256]
    const float* __restrict__ g_feats,  // [B,256]
    float* __restrict__ out)            // [B,256]
{
    __shared__ float he_t[16 * LDS_STRIDE];
    __shared__ float h1_t[16 * LDS_STRIDE];

    const int tid  = threadIdx.x;
    const int lane = tid & 31;
    const int wave = tid >> 5;          // 0..7
    const int ln   = lane & 15;         // N-column within tile / row of B
    const int hi   = lane >> 4;         // 0/1: selects K pair {0,1} vs {2,3}
    const int row0 = blockIdx.x * 16;

    // stage he tile (16x256) into LDS, coalesced
    for (int i = tid; i < 16 * 256; i += 256) {
        int r = i >> 8, c = i & 255;
        he_t[r * LDS_STRIDE + c] = he[(size_t)(row0 + r) * 256 + c];
    }
    __syncthreads();

    // ---- GEMM1: h1 = relu(he @ W1^T + b1) ---------------------------------
    #pragma unroll
    for (int t = 0; t < 2; ++t) {
        const int nt = wave + t * 8;                       // N-tile 0..15
        v8f c = {};
        const float* arow = he_t + ln * LDS_STRIDE + 2 * hi;
        const float* wrow = W1 + (size_t)(nt * 16 + ln) * 256 + 2 * hi;
        #pragma unroll 8
        for (int kb = 0; kb < 256; kb += 4) {
            v2f a  = *(const v2f*)(arow + kb);             // A[l&15][kb+2*hi+j]
            v2f bb = *(const v2f*)(wrow + kb);             // B[kb+2*hi+j][l&15] = W1[n,k]
            c = __builtin_amdgcn_wmma_f32_16x16x4_f32(
                    false, a, false, bb, (short)0, c, false, false);
        }
        const float bias = b1[nt * 16 + ln];
        #pragma unroll
        for (int r = 0; r < 8; ++r) {
            int m = r + hi * 8;                            // D row mapping
            float v = c[r] + bias;
            h1_t[m * LDS_STRIDE + nt * 16 + ln] = v > 0.f ? v : 0.f;
        }
    }
    __syncthreads();

    // ---- GEMM2: out = h1 @ W2^T + b2 + g_feats ----------------------------
    #pragma unroll
    for (int t = 0; t < 2; ++t) {
        const int nt = wave + t * 8;
        v8f c = {};
        const float* arow = h1_t + ln * LDS_STRIDE + 2 * hi;
        const float* wrow = W2 + (size_t)(nt * 16 + ln) * 256 + 2 * hi;
        #pragma unroll 8
        for (int kb = 0; kb < 256; kb += 4) {
            v2f a  = *(const v2f*)(arow + kb);
            v2f bb = *(const v2f*)(wrow + kb);
            c = __builtin_amdgcn_wmma_f32_16x16x4_f32(
                    false, a, false, bb, (short)0, c, false, false);
        }
        const float bias = b2[nt * 16 + ln];
        #pragma unroll
        for (int r = 0; r < 8; ++r) {
            int m  = r + hi * 8;
            int gr = row0 + m;
            int gc = nt * 16 + ln;
            out[(size_t)gr * 256 + gc] = c[r] + bias + g_feats[(size_t)gr * 256 + gc];
        }
    }
}

// ---------------------------------------------------------------------------
extern "C" void kernel_launch(void* const* d_in, const int* in_sizes, int n_in,
                              void* d_out, int out_size, void* d_ws, size_t ws_size,
                              hipStream_t stream) {
    const float* node_feats = (const float*)d_in[0];   // [N,256]
    const float* g_feats    = (const float*)d_in[1];   // [B,256]
    const float* degree     = (const float*)d_in[2];   // [N,1]
    const int*   seg        = (const int*)  d_in[3];   // [N]
    const float* attn       = (const float*)d_in[4];   // [1,4,64]
    const float* W1         = (const float*)d_in[5];   // [256,256]
    const float* b1         = (const float*)d_in[6];   // [256]
    const float* W2         = (const float*)d_in[7];   // [256,256]
    const float* b2         = (const float*)d_in[8];   // [256]
    float* out = (float*)d_out;

    const int N = in_sizes[0] / 256;   // 500000
    const int B = in_sizes[1] / 256;   // 4096

    float* he = (float*)d_ws;          // [B,256] scratch (4 MB)

    pool_kernel<<<B, 256, 0, stream>>>(node_feats, g_feats, degree, seg, attn, he, N);
    mlp_kernel<<<B / 16, 256, 0, stream>>>(he, W1, b1, W2, b2, g_feats, out);
}